// SampleModel_53618371723963
// MI455X (gfx1250) — compile-verified
//
#include <hip/hip_runtime.h>
#include <hip/hip_bf16.h>
#include <cstdint>

typedef __attribute__((ext_vector_type(2))) float v2f;
typedef __attribute__((ext_vector_type(8))) float v8f;

#define B_ROWS   16384
#define S_ROWS   12800
#define DDIM     128
#define LDS_PITCH (DDIM + 4)          // 132 floats = 528 B: 16B-aligned, conflict-free
#define INV_TEMP (1.0f / 0.07f)

// Issue one async global->LDS 128-bit copy (tracked by ASYNCcnt).
// lds_addr: low 32 bits of a generic pointer into __shared__ (== LDS byte addr).
__device__ __forceinline__ void async_copy_b128(void* lds_dst, const void* gsrc) {
    unsigned lds_addr = (unsigned)(size_t)lds_dst;
    asm volatile("global_load_async_to_lds_b128 %0, %1, off"
                 :: "v"(lds_addr), "v"(gsrc) : "memory");
}
__device__ __forceinline__ void wait_async0() {
    asm volatile("s_wait_asynccnt 0x0" ::: "memory");
}

// ---------------------------------------------------------------- init ----
__global__ void init_kernel(float* __restrict__ colsum,
                            unsigned long long* __restrict__ maxkey,
                            float* __restrict__ out) {
    int t = blockIdx.x * blockDim.x + threadIdx.x;
    if (t < S_ROWS) colsum[t] = 0.0f;
    if (t < B_ROWS) maxkey[t] = 0ull;     // 0 == encoded -inf
    if (t == 0)     out[0]    = 0.0f;
}

// ----------------------------------------------------------- normalize ----
__global__ __launch_bounds__(256) void normalize_kernel(
        const float* __restrict__ cent, float* __restrict__ cn) {
    const int wave = threadIdx.x >> 5;
    const int lane = threadIdx.x & 31;
    const int row  = blockIdx.x * 8 + wave;
    float4 v = ((const float4*)(cent + (size_t)row * DDIM))[lane];
    float ss = v.x * v.x + v.y * v.y + v.z * v.z + v.w * v.w;
    #pragma unroll
    for (int off = 16; off; off >>= 1) ss += __shfl_xor(ss, off, 32);
    float scale = 1.0f / fmaxf(sqrtf(ss), 1e-12f);
    float4 o = { v.x * scale, v.y * scale, v.z * scale, v.w * scale };
    ((float4*)(cn + (size_t)row * DDIM))[lane] = o;
}

// ------------------------------------------------------------- max-dot ----
// Grid (B/16, 25), 8 waves. A tile staged via async-to-LDS; each wave sweeps
// 4 centroid tiles of 16 with f32 WMMA, keeping packed (orderedF32<<32|~idx)
// running maxima, half-wave shfl reduce, then u64 atomicMax.
__global__ __launch_bounds__(256) void maxdot_kernel(
        const float* __restrict__ feat, const float* __restrict__ cn,
        unsigned long long* __restrict__ maxkey) {
    __shared__ __align__(16) float As[16][LDS_PITCH];
    const int tid  = threadIdx.x;
    const int wave = tid >> 5;
    const int lane = tid & 31;
    const int lid  = lane & 15;
    const int half = lane >> 4;
    const int m0   = blockIdx.x * 16;

    // async stage: 16 rows x 128 floats = 512 b128 transfers, 2 per thread
    #pragma unroll
    for (int i = tid; i < 512; i += 256) {
        int r = i >> 5, c4 = (i & 31) << 2;
        async_copy_b128(&As[r][c4], feat + (size_t)(m0 + r) * DDIM + c4);
    }
    wait_async0();
    __syncthreads();

    unsigned long long key[8];
    #pragma unroll
    for (int r = 0; r < 8; ++r) key[r] = 0ull;

    for (int t = 0; t < 4; ++t) {
        const int n0 = (blockIdx.y * 32 + wave * 4 + t) * 16;
        const float* brow = cn + (size_t)(n0 + lid) * DDIM;
        v8f acc = {};
        #pragma unroll 8
        for (int k0 = 0; k0 < DDIM; k0 += 4) {
            const int ks = k0 + 2 * half;          // lanes 16-31 hold K=+2,+3
            v2f a, b;
            a.x = As[lid][ks];  a.y = As[lid][ks + 1];
            b.x = brow[ks];     b.y = brow[ks + 1];
            acc = __builtin_amdgcn_wmma_f32_16x16x4_f32(
                    false, a, false, b, (short)0, acc, false, false);
        }
        const unsigned int idxenc = 0xFFFFFFFFu - (unsigned int)(n0 + lid);
        #pragma unroll
        for (int r = 0; r < 8; ++r) {
            unsigned int u = __float_as_uint(acc[r]);
            u = (u & 0x80000000u) ? ~u : (u | 0x80000000u);   // order-preserving
            unsigned long long k = ((unsigned long long)u << 32) | idxenc;
            key[r] = (k > key[r]) ? k : key[r];
        }
    }
    #pragma unroll
    for (int r = 0; r < 8; ++r) {
        #pragma unroll
        for (int off = 8; off; off >>= 1) {
            unsigned long long o = __shfl_xor(key[r], off, 32);
            key[r] = (o > key[r]) ? o : key[r];
        }
    }
    if (lid == 0) {
        #pragma unroll
        for (int r = 0; r < 8; ++r)
            atomicMax(&maxkey[m0 + half * 8 + r], key[r]);
    }
}

// -------------------------------------------------------------- colsum ----
// colsum[j] = sum_i exp(c_i . c_j / T). Grid (S/128, 50): block owns 128
// columns (16/wave), loops 16 M-tiles with DOUBLE-BUFFERED async-to-LDS A
// staging overlapped with the WMMA loop. Lane L and L+16 hold the same
// column (rows 0-7 / 8-15), so shfl_xor(16) combines before one atomicAdd.
__global__ __launch_bounds__(256) void colsum_kernel(
        const float* __restrict__ cn, float* __restrict__ colsum) {
    __shared__ __align__(16) float As[2][16][LDS_PITCH];
    const int tid  = threadIdx.x;
    const int wave = tid >> 5;
    const int lane = tid & 31;
    const int lid  = lane & 15;
    const int half = lane >> 4;
    const int n0   = blockIdx.x * 128 + wave * 16;
    const float* brow = cn + (size_t)(n0 + lid) * DDIM;

    // prologue: stage tile 0 into buffer 0
    {
        const int m0 = blockIdx.y * 256;           // (blockIdx.y*16 + 0) * 16
        #pragma unroll
        for (int i = tid; i < 512; i += 256) {
            int r = i >> 5, c4 = (i & 31) << 2;
            async_copy_b128(&As[0][r][c4], cn + (size_t)(m0 + r) * DDIM + c4);
        }
        wait_async0();
        __syncthreads();
    }

    float colacc = 0.0f;
    for (int i = 0; i < 16; ++i) {
        const int cur = i & 1, nxt = cur ^ 1;
        if (i < 15) {                              // prefetch tile i+1
            const int m0 = (blockIdx.y * 16 + i + 1) * 16;
            #pragma unroll
            for (int j = tid; j < 512; j += 256) {
                int r = j >> 5, c4 = (j & 31) << 2;
                async_copy_b128(&As[nxt][r][c4],
                                cn + (size_t)(m0 + r) * DDIM + c4);
            }
        }
        v8f acc = {};
        #pragma unroll 8
        for (int k0 = 0; k0 < DDIM; k0 += 4) {
            const int ks = k0 + 2 * half;
            v2f a, b;
            a.x = As[cur][lid][ks];  a.y = As[cur][lid][ks + 1];
            b.x = brow[ks];          b.y = brow[ks + 1];
            acc = __builtin_amdgcn_wmma_f32_16x16x4_f32(
                    false, a, false, b, (short)0, acc, false, false);
        }
        #pragma unroll
        for (int r = 0; r < 8; ++r) colacc += expf(acc[r] * INV_TEMP);

        wait_async0();                             // own copies into nxt done
        __syncthreads();                           // publish across waves
    }
    colacc += __shfl_xor(colacc, 16, 32);          // rows 0-7 + rows 8-15
    if (half == 0) atomicAdd(&colsum[n0 + lid], colacc);
}

// ---------------------------------------------------------------- loss ----
// J = log(pe) - log(pe + cs[k]) = -log1p(cs[k] * exp(-m/T));  out = mean(-J)
__global__ __launch_bounds__(256) void loss_kernel(
        const unsigned long long* __restrict__ maxkey,
        const float* __restrict__ colsum, float* __restrict__ out) {
    __shared__ float red[256];
    const int t = blockIdx.x * 256 + threadIdx.x;
    const unsigned long long k = maxkey[t];
    const unsigned int idx = 0xFFFFFFFFu - (unsigned int)(k & 0xFFFFFFFFu);
    unsigned int u = (unsigned int)(k >> 32);
    u = (u & 0x80000000u) ? (u ^ 0x80000000u) : ~u;   // decode ordered float
    const float m = __uint_as_float(u);
    const float term = log1pf(colsum[idx] * expf(-m * INV_TEMP));  // BALANCE=1
    red[threadIdx.x] = term;
    __syncthreads();
    #pragma unroll
    for (int off = 128; off; off >>= 1) {
        if (threadIdx.x < off) red[threadIdx.x] += red[threadIdx.x + off];
        __syncthreads();
    }
    if (threadIdx.x == 0) atomicAdd(out, red[0] * (1.0f / (float)B_ROWS));
}

// -------------------------------------------------------------- launch ----
extern "C" void kernel_launch(void* const* d_in, const int* in_sizes, int n_in,
                              void* d_out, int out_size, void* d_ws, size_t ws_size,
                              hipStream_t stream) {
    const float* feat = (const float*)d_in[0];   // [16384, 128] f32
    const float* cent = (const float*)d_in[1];   // [12800, 128] f32
    float* out = (float*)d_out;

    char* ws = (char*)d_ws;
    float* cn = (float*)ws;                                   // 12800*128 f32
    float* colsum = (float*)(ws + (size_t)S_ROWS * DDIM * 4); // 12800 f32
    unsigned long long* maxkey =
        (unsigned long long*)(ws + (size_t)S_ROWS * DDIM * 4 + (size_t)S_ROWS * 4);
    // maxkey offset 6,604,800 B (8B aligned); total ws use ~6.74 MB

    init_kernel<<<B_ROWS / 256, 256, 0, stream>>>(colsum, maxkey, out);
    normalize_kernel<<<S_ROWS / 8, 256, 0, stream>>>(cent, cn);
    maxdot_kernel<<<dim3(B_ROWS / 16, 25), 256, 0, stream>>>(feat, cn, maxkey);
    colsum_kernel<<<dim3(S_ROWS / 128, 50), 256, 0, stream>>>(cn, colsum);
    loss_kernel<<<B_ROWS / 256, 256, 0, stream>>>(maxkey, colsum, out);
}